// ComputeAlignmentError_84078279786857
// MI455X (gfx1250) — compile-verified
//
#include <hip/hip_runtime.h>

typedef __attribute__((ext_vector_type(2))) float v2f;
typedef __attribute__((ext_vector_type(8))) float v8f;

#define EPSF 1e-8f

struct V3 { float x, y, z; };
__device__ __forceinline__ V3 v3sub(V3 a, V3 b) { return {a.x - b.x, a.y - b.y, a.z - b.z}; }
__device__ __forceinline__ V3 v3add(V3 a, V3 b) { return {a.x + b.x, a.y + b.y, a.z + b.z}; }
__device__ __forceinline__ float v3dot(V3 a, V3 b) { return a.x * b.x + a.y * b.y + a.z * b.z; }
__device__ __forceinline__ V3 v3nrm(V3 a) {
    // matches F.normalize: x / max(||x||, eps)
    float inv = 1.0f / fmaxf(sqrtf(v3dot(a, a)), EPSF);
    return {a.x * inv, a.y * inv, a.z * inv};
}
__device__ __forceinline__ V3 v3cross(V3 a, V3 b) {
    return {a.y * b.z - a.z * b.y, a.z * b.x - a.x * b.z, a.x * b.y - a.y * b.x};
}

__device__ __forceinline__ void frame_basis(const float* f, V3& origin, V3 E[3]) {
    // frames[..., d, col]: col 0=a, 1=b, 2=c ; component d has stride 3
    V3 pa = {f[0], f[3], f[6]};
    V3 pb = {f[1], f[4], f[7]};
    V3 pc = {f[2], f[5], f[8]};
    V3 w1 = v3nrm(v3sub(pa, pb));
    V3 w2 = v3nrm(v3sub(pc, pb));
    E[0] = v3nrm(v3add(w1, w2));
    E[1] = v3nrm(v3sub(w2, w1));
    E[2] = v3cross(E[0], E[1]);
    origin = pb;
}

// Kernel 1: per-(batch, index) prep. Builds:
//   A    [b][n][8]     = [p, q, 0, 0]          (GEMM A rows, K padded to 8)
//   W    [b][3][8][n]  = [Ep_k, -Et_k, 0, 0]   (GEMM B in K x N row-major, per k)
//   Bias [b][3][n]     = ct_k - cp_k + eps
__global__ void prep_kernel(const float* __restrict__ pcoord, const float* __restrict__ tcoord,
                            const float* __restrict__ pframe, const float* __restrict__ tframe,
                            float* __restrict__ A, float* __restrict__ W,
                            float* __restrict__ Bias, int n, int total) {
    int idx = blockIdx.x * blockDim.x + threadIdx.x;
    if (idx >= total) return;
    int bb = idx / n;
    int j  = idx - bb * n;

    V3 op, Ep[3], ot, Et[3];
    frame_basis(pframe + (size_t)idx * 9, op, Ep);
    frame_basis(tframe + (size_t)idx * 9, ot, Et);

    const float* p = pcoord + (size_t)idx * 3;
    const float* q = tcoord + (size_t)idx * 3;
    float* Arow = A + (size_t)idx * 8;
    Arow[0] = p[0]; Arow[1] = p[1]; Arow[2] = p[2];
    Arow[3] = q[0]; Arow[4] = q[1]; Arow[5] = q[2];
    Arow[6] = 0.0f; Arow[7] = 0.0f;

#pragma unroll
    for (int k = 0; k < 3; ++k) {
        float* Wk = W + (size_t)(bb * 3 + k) * 8 * n;
        Wk[(size_t)0 * n + j] = Ep[k].x;
        Wk[(size_t)1 * n + j] = Ep[k].y;
        Wk[(size_t)2 * n + j] = Ep[k].z;
        Wk[(size_t)3 * n + j] = -Et[k].x;
        Wk[(size_t)4 * n + j] = -Et[k].y;
        Wk[(size_t)5 * n + j] = -Et[k].z;
        Wk[(size_t)6 * n + j] = 0.0f;
        Wk[(size_t)7 * n + j] = 0.0f;
        Bias[(size_t)(bb * 3 + k) * n + j] = v3dot(ot, Et[k]) - v3dot(op, Ep[k]) + EPSF;
    }
}

// Kernel 2: one wave per 16x16 (i,j) output tile.
// D_k = A(16x8) * W_k(8x16) + bias  via two chained V_WMMA_F32_16X16X4_F32,
// epilogue err = sqrt(d0^2+d1^2+d2^2) * rowmask * colmask.
__global__ __launch_bounds__(256) void wmma_err_kernel(
    const float* __restrict__ A, const float* __restrict__ W,
    const float* __restrict__ Bias, const unsigned char* __restrict__ mask,
    float* __restrict__ out, int n, int nt, int tiles) {
    int wave = threadIdx.x >> 5;
    int lane = threadIdx.x & 31;
    int tid  = blockIdx.x * (blockDim.x >> 5) + wave;
    if (tid >= tiles) return;  // wave-uniform: EXEC stays all-ones below

    int bb  = tid / (nt * nt);
    int rem = tid - bb * nt * nt;
    int it  = rem / nt;
    int jt  = rem - it * nt;
    int ib  = it << 4, jb = jt << 4;

    int col  = lane & 15;
    int half = lane >> 4;

    // A fragment: lane holds row M = col; VGPR v holds K = v + 2*half.
    const float* Arow = A + ((size_t)(bb * n + ib + col)) * 8;
    v2f a0 = {Arow[2 * half],     Arow[2 * half + 1]};      // K 0..3
    v2f a1 = {Arow[4 + 2 * half], Arow[4 + 2 * half + 1]};  // K 4..7

    int jcol = jb + col;
    v8f d0, d1, d2;
#pragma unroll
    for (int k = 0; k < 3; ++k) {
        const float* Wk = W + (size_t)(bb * 3 + k) * 8 * n;
        // B fragment: lane holds col N = col; VGPR v holds K = v + 2*half.
        int r0 = 2 * half * n + jcol;       // K row 0 chunk (kk = 2*half)
        int r1 = (4 + 2 * half) * n + jcol; // K row 0 of second chunk (kk = 4+2*half)
        v2f b0 = {Wk[r0], Wk[r0 + n]};
        v2f b1 = {Wk[r1], Wk[r1 + n]};

        float bias = Bias[(size_t)(bb * 3 + k) * n + jcol];
        v8f c;
#pragma unroll
        for (int r = 0; r < 8; ++r) c[r] = bias;  // C[row][N=col] seeded with bias_{j,k}

        // 8-arg form: (neg_a, A, neg_b, B, c_mod, C, reuse_a, reuse_b)
        c = __builtin_amdgcn_wmma_f32_16x16x4_f32(false, a0, false, b0, (short)0, c, false, false);
        c = __builtin_amdgcn_wmma_f32_16x16x4_f32(false, a1, false, b1, (short)0, c, false, false);
        if (k == 0) d0 = c; else if (k == 1) d1 = c; else d2 = c;
    }

    float colm = (float)mask[bb * n + jcol];
    float rowm_src = (float)mask[bb * n + ib + col];  // lane L holds mask of row L%16

    float* outrow0 = out + (size_t)(bb * n + ib + 8 * half) * n + jcol;
#pragma unroll
    for (int r = 0; r < 8; ++r) {
        float rowm = __shfl(rowm_src, r + 8 * half, 32);
        float s = d0[r] * d0[r] + d1[r] * d1[r] + d2[r] * d2[r];
        // raw v_sqrt_f32 (~1 ULP): args are sums of squares, no denormal rescale needed
        float e = __builtin_amdgcn_sqrtf(s);
        e *= rowm * colm;
        outrow0[(size_t)r * n] = e;
    }
}

extern "C" void kernel_launch(void* const* d_in, const int* in_sizes, int n_in,
                              void* d_out, int out_size, void* d_ws, size_t ws_size,
                              hipStream_t stream) {
    const float* pcoord = (const float*)d_in[0];
    const float* tcoord = (const float*)d_in[1];
    const float* pframe = (const float*)d_in[2];
    const float* tframe = (const float*)d_in[3];
    const unsigned char* mask = (const unsigned char*)d_in[4];

    int bn = in_sizes[4];        // b*n
    int n  = out_size / bn;      // out = b*n*n
    int b  = bn / n;

    float* A    = (float*)d_ws;                    // b*n*8
    float* W    = A + (size_t)b * n * 8;           // b*3*8*n
    float* Bias = W + (size_t)b * 3 * 8 * n;       // b*3*n

    int total = b * n;
    prep_kernel<<<(total + 255) / 256, 256, 0, stream>>>(pcoord, tcoord, pframe, tframe,
                                                         A, W, Bias, n, total);

    int nt = n / 16;
    int tiles = b * nt * nt;
    int wavesPerBlock = 8;  // 256 threads = 8 wave32
    int blocks = (tiles + wavesPerBlock - 1) / wavesPerBlock;
    wmma_err_kernel<<<blocks, 256, 0, stream>>>(A, W, Bias, mask, (float*)d_out, n, nt, tiles);
}